// RotatEHead_68599217652387
// MI455X (gfx1250) — compile-verified
//
#include <hip/hip_runtime.h>
#include <math.h>

#define HALF       256                    // dim/2
#define DIM        512
#define NUM_REL    64
#define TAB_FLOATS (NUM_REL * HALF * 2)   // cos table then sin table: 32768 floats
#define TAB_BYTES  (TAB_FLOATS * (int)sizeof(float))  // 131072 bytes
#define BLOCK      256
#define WPB        (BLOCK / 32)           // waves (edges) per block
#define MAX_GRID   2048

#define AS1 __attribute__((address_space(1)))
#define AS3 __attribute__((address_space(3)))

typedef int v4i __attribute__((ext_vector_type(4)));

static __device__ __forceinline__ float term(float rh, float ih, float rt, float it,
                                             float c, float s) {
    // re_diff = rh*c - ih*s - rt ; im_diff = rh*s + ih*c - it
    float rd = fmaf(rh, c, -fmaf(ih, s, rt));
    float id = fmaf(rh, s, fmaf(ih, c, -it));
    return sqrtf(fmaf(rd, rd, id * id));
}

static __device__ __forceinline__ float sum4(float4 rh, float4 ih, float4 rt, float4 it,
                                             float4 c, float4 s) {
    return term(rh.x, ih.x, rt.x, it.x, c.x, s.x)
         + term(rh.y, ih.y, rt.y, it.y, c.y, s.y)
         + term(rh.z, ih.z, rt.z, it.z, c.z, s.z)
         + term(rh.w, ih.w, rt.w, it.w, c.w, s.w);
}

// ---- Main pass: persistent blocks; async-stage table to LDS; one wave per edge ----
__global__ __launch_bounds__(BLOCK) void rotate_score_kernel(
    const float* __restrict__ node, const int* __restrict__ eidx,
    const int* __restrict__ rtype, const float* __restrict__ tab,
    const float* __restrict__ tptr, const float* __restrict__ bptr,
    float* __restrict__ out, int E) {
    extern __shared__ float smem[];  // TAB_BYTES of dynamic LDS

#if __has_builtin(__builtin_amdgcn_global_load_async_to_lds_b128)
    // CDNA5 async global->LDS DMA path (tracked by ASYNCcnt)
    #pragma unroll
    for (int i = 0; i < TAB_FLOATS / (BLOCK * 4); ++i) {     // 32 iters of B128/lane
        int off = (i * BLOCK + threadIdx.x) * 4;             // float index
        __builtin_amdgcn_global_load_async_to_lds_b128(
            (AS1 v4i*)(tab + off), (AS3 v4i*)(smem + off), 0, 0);
    }
  #if __has_builtin(__builtin_amdgcn_s_wait_asynccnt)
    __builtin_amdgcn_s_wait_asynccnt(0);
  #else
    asm volatile("s_wait_asynccnt 0" ::: "memory");
  #endif
#else
    #pragma unroll
    for (int i = 0; i < TAB_FLOATS / (BLOCK * 4); ++i) {
        int off = i * BLOCK + threadIdx.x;
        ((float4*)smem)[off] = ((const float4*)tab)[off];
    }
#endif
    __syncthreads();

    const float temp  = tptr[0];
    const float bias  = bptr[0];
    const float scale = -1.0f / (temp * (float)HALF);  // out = bias - mean/temp

    const int lane   = threadIdx.x & 31;
    const int wid    = blockIdx.x * WPB + (threadIdx.x >> 5);
    const int stride = gridDim.x * WPB;
    const float4* smem4 = (const float4*)smem;

    for (int e = wid; e < E; e += stride) {
        // e is wave-uniform: force index fetches onto the scalar path (s_load/KMcnt)
        int eu = __builtin_amdgcn_readfirstlane(e);
        int h  = __builtin_amdgcn_readfirstlane(eidx[eu]);
        int t  = __builtin_amdgcn_readfirstlane(eidx[E + eu]);
        int r  = __builtin_amdgcn_readfirstlane(rtype[eu]);

        const float4* nh = (const float4*)(node + (size_t)h * DIM);
        const float4* nt = (const float4*)(node + (size_t)t * DIM);

        // Prefetch next iteration's gather rows (one 64B line per lane -> full 2KB row)
        int en = e + stride;
        if (en < E) {
            int enu = __builtin_amdgcn_readfirstlane(en);
            int hn  = __builtin_amdgcn_readfirstlane(eidx[enu]);
            int tn  = __builtin_amdgcn_readfirstlane(eidx[E + enu]);
            const char* ph = (const char*)(node + (size_t)hn * DIM);
            const char* pt = (const char*)(node + (size_t)tn * DIM);
            __builtin_prefetch(ph + lane * 64, 0, 3);
            __builtin_prefetch(pt + lane * 64, 0, 3);
        }

        // lane owns complex components k in {4l..4l+3} and {128+4l..128+4l+3}
        int cb = r * (HALF / 4);  // float4 index of this relation's cos row
        float4 c0 = smem4[cb + lane];
        float4 c1 = smem4[cb + lane + 32];
        float4 s0 = smem4[(TAB_FLOATS / 8) + cb + lane];
        float4 s1 = smem4[(TAB_FLOATS / 8) + cb + lane + 32];

        float4 reh0 = nh[lane],      reh1 = nh[lane + 32];
        float4 imh0 = nh[lane + 64], imh1 = nh[lane + 96];
        float4 ret0 = nt[lane],      ret1 = nt[lane + 32];
        float4 imt0 = nt[lane + 64], imt1 = nt[lane + 96];

        float sum = sum4(reh0, imh0, ret0, imt0, c0, s0)
                  + sum4(reh1, imh1, ret1, imt1, c1, s1);

        #pragma unroll
        for (int m = 16; m; m >>= 1) sum += __shfl_xor(sum, m, 32);

        if (lane == 0) __builtin_nontemporal_store(fmaf(sum, scale, bias), &out[e]);
    }
}

// ---- Pass 1: tiny kernel, precompute cos/sin(phase) table into workspace ----
__global__ __launch_bounds__(BLOCK) void rotate_build_table_kernel(
    const float* __restrict__ rel, float* __restrict__ tab) {
    int i = blockIdx.x * blockDim.x + threadIdx.x;
    if (i < NUM_REL * HALF) {
        float ph = rel[i] * (3.14159265358979323846f / 9.0f);  // rel / (MARGIN/PI)
        float s, c;
        sincosf(ph, &s, &c);
        tab[i] = c;                     // cos region [0, 16384)
        tab[NUM_REL * HALF + i] = s;    // sin region [16384, 32768)
    }
}

// ---- Fallback if workspace is too small: compute sin/cos inline ----
__global__ __launch_bounds__(BLOCK) void rotate_score_inline_kernel(
    const float* __restrict__ node, const int* __restrict__ eidx,
    const int* __restrict__ rtype, const float* __restrict__ rel,
    const float* __restrict__ tptr, const float* __restrict__ bptr,
    float* __restrict__ out, int E) {
    const float temp  = tptr[0];
    const float bias  = bptr[0];
    const float scale = -1.0f / (temp * (float)HALF);
    const float K     = 3.14159265358979323846f / 9.0f;

    const int lane   = threadIdx.x & 31;
    const int wid    = blockIdx.x * WPB + (threadIdx.x >> 5);
    const int stride = gridDim.x * WPB;

    for (int e = wid; e < E; e += stride) {
        int eu = __builtin_amdgcn_readfirstlane(e);
        int h  = __builtin_amdgcn_readfirstlane(eidx[eu]);
        int t  = __builtin_amdgcn_readfirstlane(eidx[E + eu]);
        int r  = __builtin_amdgcn_readfirstlane(rtype[eu]);
        const float4* nh = (const float4*)(node + (size_t)h * DIM);
        const float4* nt = (const float4*)(node + (size_t)t * DIM);
        const float4* rr = (const float4*)(rel + (size_t)r * HALF);

        float4 p0 = rr[lane], p1 = rr[lane + 32];
        float4 c0, s0, c1, s1;
        sincosf(p0.x * K, &s0.x, &c0.x); sincosf(p0.y * K, &s0.y, &c0.y);
        sincosf(p0.z * K, &s0.z, &c0.z); sincosf(p0.w * K, &s0.w, &c0.w);
        sincosf(p1.x * K, &s1.x, &c1.x); sincosf(p1.y * K, &s1.y, &c1.y);
        sincosf(p1.z * K, &s1.z, &c1.z); sincosf(p1.w * K, &s1.w, &c1.w);

        float4 reh0 = nh[lane],      reh1 = nh[lane + 32];
        float4 imh0 = nh[lane + 64], imh1 = nh[lane + 96];
        float4 ret0 = nt[lane],      ret1 = nt[lane + 32];
        float4 imt0 = nt[lane + 64], imt1 = nt[lane + 96];

        float sum = sum4(reh0, imh0, ret0, imt0, c0, s0)
                  + sum4(reh1, imh1, ret1, imt1, c1, s1);

        #pragma unroll
        for (int m = 16; m; m >>= 1) sum += __shfl_xor(sum, m, 32);

        if (lane == 0) __builtin_nontemporal_store(fmaf(sum, scale, bias), &out[e]);
    }
}

extern "C" void kernel_launch(void* const* d_in, const int* in_sizes, int n_in,
                              void* d_out, int out_size, void* d_ws, size_t ws_size,
                              hipStream_t stream) {
    const float* node = (const float*)d_in[0];   // (100000, 512) f32
    const int*   eidx = (const int*)d_in[1];     // (2, E) int
    const int*   rtyp = (const int*)d_in[2];     // (E,) int
    const float* rel  = (const float*)d_in[3];   // (64, 256) f32
    const float* tpt  = (const float*)d_in[4];   // scalar
    const float* bpt  = (const float*)d_in[5];   // scalar
    float* out = (float*)d_out;                  // (E,) f32
    const int E = in_sizes[2];

    if (ws_size >= (size_t)TAB_BYTES) {
        rotate_build_table_kernel<<<(NUM_REL * HALF + BLOCK - 1) / BLOCK, BLOCK, 0, stream>>>(
            rel, (float*)d_ws);
        int grid = (E + WPB - 1) / WPB;
        if (grid > MAX_GRID) grid = MAX_GRID;   // persistent blocks; table staged once each
        rotate_score_kernel<<<grid, BLOCK, TAB_BYTES, stream>>>(
            node, eidx, rtyp, (const float*)d_ws, tpt, bpt, out, E);
    } else {
        int grid = (E + WPB - 1) / WPB;
        if (grid > 8 * MAX_GRID) grid = 8 * MAX_GRID;
        rotate_score_inline_kernel<<<grid, BLOCK, 0, stream>>>(
            node, eidx, rtyp, rel, tpt, bpt, out, E);
    }
}